// entropy_loss_33500744909423
// MI455X (gfx1250) — compile-verified
//
#include <hip/hip_runtime.h>

// ---------------------------------------------------------------------------
// entropy loss: col-L2-normalize (dim=0), then mean over rows of
//   -sum_j p_ij * log(p_ij + 1e-8)
//
// Memory-bound two-pass streaming reduction (min ~320MB HBM traffic with the
// reversed second pass exploiting the 192MB L2). WMMA (v_wmma_f32_16x16x4_f32)
// performs the wave32 cross-lane fold; ds_swizzle finishes the 16-way sum.
// ---------------------------------------------------------------------------

typedef __attribute__((ext_vector_type(2))) float v2f;
typedef __attribute__((ext_vector_type(8))) float v8f;

#define ROWS   16384
#define COLS   4096
#define COLS4  (COLS / 4)
#define NPART  2048          // entropy block partials

// Wave32 all-lane sum. Requires EXEC == all ones (callers guarantee uniform
// control flow). One WMMA folds 32 lanes -> 16 distinct values:
//   A = ones(16x4), B: b0 = per-lane value, b1 = 0
//   => D[m][n] = v_n + v_{n+16} for every m; d[0] holds that value in every
//      lane keyed by (lane & 15). Four XOR ds_swizzles finish the sum.
__device__ __forceinline__ float wave_reduce32(float v) {
    v2f a; a[0] = 1.0f; a[1] = 1.0f;   // A-matrix: all ones
    v2f b; b[0] = v;    b[1] = 0.0f;   // B-matrix: lane values in one K-plane
    v8f c = {};
    c = __builtin_amdgcn_wmma_f32_16x16x4_f32(
            /*neg_a=*/false, a, /*neg_b=*/false, b,
            /*c_mod=*/(short)0, c, /*reuse_a=*/false, /*reuse_b=*/false);
    float s = c[0];                    // = v_{lane&15} + v_{(lane&15)+16}
    // group-of-32 XOR swizzles: offset = (xor_mask<<10) | 0x1F (and_mask)
    s += __int_as_float(__builtin_amdgcn_ds_swizzle(__float_as_int(s), 0x201F)); // xor 8
    s += __int_as_float(__builtin_amdgcn_ds_swizzle(__float_as_int(s), 0x101F)); // xor 4
    s += __int_as_float(__builtin_amdgcn_ds_swizzle(__float_as_int(s), 0x081F)); // xor 2
    s += __int_as_float(__builtin_amdgcn_ds_swizzle(__float_as_int(s), 0x041F)); // xor 1
    return s;                          // total over all 32 lanes, in all lanes
}

// Block (256-thread) sum -> returned valid in thread 0.
__device__ __forceinline__ float block_reduce256(float v, float* lds8) {
    float w = wave_reduce32(v);
    int wid = threadIdx.x >> 5;
    if ((threadIdx.x & 31) == 0) lds8[wid] = w;
    __syncthreads();
    float t = 0.0f;
    if (threadIdx.x == 0) {
#pragma unroll
        for (int i = 0; i < 8; ++i) t += lds8[i];
    }
    return t;
}

__global__ void init_kernel(float* __restrict__ colsum) {
    int i = blockIdx.x * blockDim.x + threadIdx.x;
    if (i < COLS) colsum[i] = 0.0f;
}

// Pass 1: colsum[c] = sum_r x[r][c]^2.
// grid = (COLS4/256, 256), block = 256. Each thread owns one float4 column
// group within a 64-row strip: fully coalesced 16B/lane loads. Atomic adds
// are spread over 4096 addresses (~256 per address) -> no hot spot.
__global__ void colsumsq_kernel(const float* __restrict__ x,
                                float* __restrict__ colsum) {
    const int g  = blockIdx.x * blockDim.x + threadIdx.x;   // float4 col group
    const int rows_per = ROWS / gridDim.y;
    const int r0 = blockIdx.y * rows_per;
    const float4* __restrict__ x4 = (const float4*)x;
    float4 acc = make_float4(0.f, 0.f, 0.f, 0.f);
    for (int r = r0; r < r0 + rows_per; ++r) {
        float4 v = x4[(size_t)r * COLS4 + g];
        acc.x += v.x * v.x;
        acc.y += v.y * v.y;
        acc.z += v.z * v.z;
        acc.w += v.w * v.w;
    }
    float* cs = colsum + g * 4;
    atomicAdd(cs + 0, acc.x);
    atomicAdd(cs + 1, acc.y);
    atomicAdd(cs + 2, acc.z);
    atomicAdd(cs + 3, acc.w);
}

__global__ void invnorm_kernel(const float* __restrict__ colsum,
                               float* __restrict__ inv) {
    int j = blockIdx.x * blockDim.x + threadIdx.x;
    if (j < COLS) {
        float n = sqrtf(colsum[j]);
        inv[j] = 1.0f / fmaxf(n, 1e-12f);
    }
}

// Pass 2: partial[b] = block's share of sum_ij p_ij * log(p_ij + 1e-8).
// REVERSED traversal: pass 1 leaves the most-recently-streamed ~192MB (the
// matrix tail) resident in L2; reading back-to-front harvests those hits
// before they are evicted, cutting pass-2 HBM traffic by up to ~75%.
// Uniform trip count (exactly 32 iters/thread) so EXEC is all-ones at WMMA.
__global__ void entropy_kernel(const float* __restrict__ x,
                               const float* __restrict__ inv,
                               float* __restrict__ partial) {
    const float4* __restrict__ x4   = (const float4*)x;
    const float4* __restrict__ inv4 = (const float4*)inv;
    const size_t n4     = (size_t)ROWS * COLS4;
    const size_t stride = (size_t)gridDim.x * blockDim.x;
    const int    iters  = (int)(n4 / stride);   // launch chosen so this is exact
    size_t i = (size_t)blockIdx.x * blockDim.x + threadIdx.x;
    float acc = 0.0f;
    for (int it = 0; it < iters; ++it, i += stride) {
        size_t j  = n4 - 1 - i;                  // back-to-front
        int    cg = (int)(j & (size_t)(COLS4 - 1));
        float4 v = x4[j];
        float4 w = inv4[cg];
        float p0 = v.x * w.x, p1 = v.y * w.y, p2 = v.z * w.z, p3 = v.w * w.w;
        acc += p0 * __logf(p0 + 1e-8f);
        acc += p1 * __logf(p1 + 1e-8f);
        acc += p2 * __logf(p2 + 1e-8f);
        acc += p3 * __logf(p3 + 1e-8f);
    }
    __shared__ float lds8[8];
    float bsum = block_reduce256(acc, lds8);
    if (threadIdx.x == 0) partial[blockIdx.x] = bsum;
}

// Final: out = -(1/ROWS) * sum(partial).  One block, no atomics anywhere.
__global__ void finalize_kernel(const float* __restrict__ partial,
                                float* __restrict__ out) {
    float acc = 0.0f;
#pragma unroll
    for (int i = threadIdx.x; i < NPART; i += 256) acc += partial[i];
    __shared__ float lds8[8];
    float t = block_reduce256(acc, lds8);
    if (threadIdx.x == 0) out[0] = t * (-1.0f / (float)ROWS);
}

extern "C" void kernel_launch(void* const* d_in, const int* in_sizes, int n_in,
                              void* d_out, int out_size, void* d_ws, size_t ws_size,
                              hipStream_t stream) {
    (void)in_sizes; (void)n_in; (void)out_size; (void)ws_size;
    const float* x = (const float*)d_in[0];
    float* out     = (float*)d_out;
    float* colsum  = (float*)d_ws;          // 4096 floats
    float* inv     = colsum + COLS;         // 4096 floats
    float* partial = inv + COLS;            // NPART floats

    init_kernel    <<<dim3((COLS + 255) / 256), dim3(256), 0, stream>>>(colsum);
    colsumsq_kernel<<<dim3(COLS4 / 256, 256),   dim3(256), 0, stream>>>(x, colsum);
    invnorm_kernel <<<dim3(COLS / 256),         dim3(256), 0, stream>>>(colsum, inv);
    // n4 = 16,777,216 float4s; 2048*256 threads -> exactly 32 iterations/thread
    entropy_kernel <<<dim3(NPART),              dim3(256), 0, stream>>>(x, inv, partial);
    finalize_kernel<<<dim3(1),                  dim3(256), 0, stream>>>(partial, out);
}